// MaskExtractor_47038481826287
// MI455X (gfx1250) — compile-verified
//
#include <hip/hip_runtime.h>
#include <math.h>

#define MNUM 8
#define FHH 24
#define FWW 24
#define CDIM 1024
#define MHH 112
#define MWW 112
#define NTOK (MHH*MWW)   // 12544
#define KCENT 16
#define HIDD 4096
#define KM_ITERS 5

typedef float v2f __attribute__((ext_vector_type(2)));
typedef float v8f __attribute__((ext_vector_type(8)));

__device__ __forceinline__ v8f wmma_f32x4(v2f a, v2f b, v8f c) {
  // V_WMMA_F32_16X16X4_F32 : D = A(16x4) x B(4x16) + C(16x16), all f32
  return __builtin_amdgcn_wmma_f32_16x16x4_f32(false, a, false, b, (short)0, c,
                                               false, false);
}

// ---------------------------------------------------------------------------
// x = feat + pos @ Wp.T + bp      (per mask; 19MB, stays in L2)
// ---------------------------------------------------------------------------
__global__ __launch_bounds__(256) void k_compute_x(
    const float* __restrict__ feats, const float* __restrict__ box_xy,
    const float* __restrict__ raw_hw, const float* __restrict__ Wp,
    const float* __restrict__ bp, float* __restrict__ X) {
  int b = blockIdx.x;
  int m = b / (FHH * FWW);
  int rem = b % (FHH * FWW);
  int h = rem / FWW, w = rem % FWW;
  float bx0 = box_xy[m * 4 + 0], by0 = box_xy[m * 4 + 1];
  float bx1 = box_xy[m * 4 + 2], by1 = box_xy[m * 4 + 3];
  float rh = raw_hw[m * 2 + 0], rw = raw_hw[m * 2 + 1];
  float posx = ((float)w / (float)FWW * (bx1 - bx0) + bx0) / (rw - 1.0f);
  float posy = ((float)h / (float)FHH * (by1 - by0) + by0) / (rh - 1.0f);
  size_t base = ((size_t)m * FHH * FWW + rem) * CDIM;
  int c = threadIdx.x * 4;  // 256 threads * 4 = 1024
  float4 f = *(const float4*)(feats + base + c);
  v2f wp0 = *(const v2f*)(Wp + (size_t)(c + 0) * 2);
  v2f wp1 = *(const v2f*)(Wp + (size_t)(c + 1) * 2);
  v2f wp2 = *(const v2f*)(Wp + (size_t)(c + 2) * 2);
  v2f wp3 = *(const v2f*)(Wp + (size_t)(c + 3) * 2);
  float4 o;
  o.x = f.x + posx * wp0.x + posy * wp0.y + bp[c + 0];
  o.y = f.y + posx * wp1.x + posy * wp1.y + bp[c + 1];
  o.z = f.z + posx * wp2.x + posy * wp2.y + bp[c + 2];
  o.w = f.w + posx * wp3.x + posy * wp3.y + bp[c + 3];
  *(float4*)(X + base + c) = o;
}

// ---------------------------------------------------------------------------
// Bilinear geometry per token (mask-independent) + per-mask 0/1 weights
// jax.image.resize 'bilinear': in = (o+0.5)*in/out - 0.5, edge-clamped
// ---------------------------------------------------------------------------
__global__ __launch_bounds__(256) void k_geo_w(const float* __restrict__ masks,
                                               int4* __restrict__ geo,
                                               float* __restrict__ Wt) {
  int t = blockIdx.x * blockDim.x + threadIdx.x;
  if (t >= NTOK) return;
  int oy = t / MWW, ox = t % MWW;
  float cy = (oy + 0.5f) * ((float)FHH / (float)MHH) - 0.5f;
  float cx = (ox + 0.5f) * ((float)FWW / (float)MWW) - 0.5f;
  float fy0 = floorf(cy), fx0 = floorf(cx);
  float fy = cy - fy0, fx = cx - fx0;
  int y0 = (int)fy0, x0 = (int)fx0;
  int y0c = min(max(y0, 0), FHH - 1), y1c = min(max(y0 + 1, 0), FHH - 1);
  int x0c = min(max(x0, 0), FWW - 1), x1c = min(max(x0 + 1, 0), FWW - 1);
  int4 g;
  g.x = y0c | (y1c << 8) | (x0c << 16) | (x1c << 24);
  g.y = __float_as_int(fy);
  g.z = __float_as_int(fx);
  g.w = 0;
  geo[t] = g;
  for (int m = 0; m < MNUM; m++)
    Wt[(size_t)m * NTOK + t] = (masks[(size_t)m * NTOK + t] > 0.0f) ? 1.0f : 0.0f;
}

__device__ __forceinline__ void unpack_geo(int4 g, int& y0, int& y1, int& x0,
                                           int& x1, float& w00, float& w01,
                                           float& w10, float& w11) {
  y0 = g.x & 0xff; y1 = (g.x >> 8) & 0xff;
  x0 = (g.x >> 16) & 0xff; x1 = (g.x >> 24) & 0xff;
  float fy = __int_as_float(g.y), fx = __int_as_float(g.z);
  w00 = (1.0f - fy) * (1.0f - fx); w01 = (1.0f - fy) * fx;
  w10 = fy * (1.0f - fx); w11 = fy * fx;
}

// ---------------------------------------------------------------------------
// Initial centroid indices: first 16 flat indices with mask>0 (stable argsort
// of 1-w), padded with first non-positive indices. One wave32 per mask.
// ---------------------------------------------------------------------------
__global__ __launch_bounds__(32) void k_init_select(const float* __restrict__ Wt,
                                                    int* __restrict__ idxOut) {
  __shared__ int shPos[16];
  __shared__ int shNeg[16];
  int m = blockIdx.x;
  int lane = threadIdx.x;
  unsigned lowmask = (lane == 0) ? 0u : ((1u << lane) - 1u);
  int posCnt = 0, negCnt = 0;
  for (int base = 0; base < NTOK; base += 32) {
    float wv = Wt[(size_t)m * NTOK + base + lane];
    bool p = wv > 0.0f;
    unsigned bal = (unsigned)__ballot(p);
    int r = posCnt + __popc(bal & lowmask);
    if (p && r < 16) shPos[r] = base + lane;
    unsigned nbal = ~bal;
    int nr = negCnt + __popc(nbal & lowmask);
    if (!p && nr < 16) shNeg[nr] = base + lane;
    posCnt += __popc(bal);
    negCnt += __popc(nbal);
    if (posCnt >= 16 && negCnt >= 16) break;
  }
  __syncthreads();
  if (lane < 16) {
    int pc = posCnt < 16 ? posCnt : 16;
    idxOut[m * 16 + lane] = (lane < pc) ? shPos[lane] : shNeg[lane - pc];
  }
}

// ---------------------------------------------------------------------------
// Gather initial centroids (bilinear sample of X at selected token positions)
// ---------------------------------------------------------------------------
__global__ __launch_bounds__(256) void k_gather_cent(
    const float* __restrict__ X, const int4* __restrict__ geo,
    const int* __restrict__ idxIn, float* __restrict__ cent) {
  int m = blockIdx.x / KCENT, j = blockIdx.x % KCENT;
  int t = idxIn[m * KCENT + j];
  int y0, y1, x0, x1; float w00, w01, w10, w11;
  unpack_geo(geo[t], y0, y1, x0, x1, w00, w01, w10, w11);
  const float* xm = X + (size_t)m * FHH * FWW * CDIM;
  const float* p00 = xm + (size_t)(y0 * FWW + x0) * CDIM;
  const float* p01 = xm + (size_t)(y0 * FWW + x1) * CDIM;
  const float* p10 = xm + (size_t)(y1 * FWW + x0) * CDIM;
  const float* p11 = xm + (size_t)(y1 * FWW + x1) * CDIM;
  float* out = cent + ((size_t)m * KCENT + j) * CDIM;
  for (int c = threadIdx.x; c < CDIM; c += 256)
    out[c] = w00 * p00[c] + w01 * p01[c] + w10 * p10[c] + w11 * p11[c];
}

// ---------------------------------------------------------------------------
// Assignment: one wave per 16-token tile. WMMA f32 16x16x4 computes
// dots(16 tokens x 16 centroids) over K=1024; argmax(dot - 0.5*cq) = argmin d2
// ---------------------------------------------------------------------------
__global__ __launch_bounds__(256) void k_assign(const float* __restrict__ X,
                                                const float* __restrict__ centIn,
                                                const int4* __restrict__ geo,
                                                int* __restrict__ labels) {
  int wave = threadIdx.x >> 5, lane = threadIdx.x & 31;
  int m = blockIdx.x / 98, tg = blockIdx.x % 98;
  int tile = tg * 8 + wave;
  int base = tile * 16;
  int Mrow = lane & 15;
  int tok = base + Mrow;
  int y0, y1, x0, x1; float w00, w01, w10, w11;
  unpack_geo(geo[tok], y0, y1, x0, x1, w00, w01, w10, w11);
  const float* xm = X + (size_t)m * FHH * FWW * CDIM;
  const float* p00 = xm + (size_t)(y0 * FWW + x0) * CDIM;
  const float* p01 = xm + (size_t)(y0 * FWW + x1) * CDIM;
  const float* p10 = xm + (size_t)(y1 * FWW + x0) * CDIM;
  const float* p11 = xm + (size_t)(y1 * FWW + x1) * CDIM;
  const float* cb = centIn + ((size_t)m * KCENT + Mrow) * CDIM;
  int cOff = (lane < 16) ? 0 : 2;
  v8f acc = {0.f, 0.f, 0.f, 0.f, 0.f, 0.f, 0.f, 0.f};
  float cq = 0.0f;
  for (int k0 = 0; k0 < CDIM; k0 += 4) {
    int c0 = k0 + cOff;
    v2f q00 = *(const v2f*)(p00 + c0);
    v2f q01 = *(const v2f*)(p01 + c0);
    v2f q10 = *(const v2f*)(p10 + c0);
    v2f q11 = *(const v2f*)(p11 + c0);
    v2f a = q00 * w00 + q01 * w01 + q10 * w10 + q11 * w11;
    v2f bv = *(const v2f*)(cb + c0);
    cq += bv.x * bv.x + bv.y * bv.y;
    acc = wmma_f32x4(a, bv, acc);
  }
  cq += __shfl_xor(cq, 16);  // lane L and L+16 cover complementary K of cent L
#pragma unroll
  for (int v = 0; v < 8; v++) {
    float val = acc[v] - 0.5f * cq;
    int idx = Mrow;
#pragma unroll
    for (int d = 1; d < 16; d <<= 1) {  // reduce within each 16-lane half
      float ov = __shfl_xor(val, d);
      int oi = __shfl_xor(idx, d);
      bool take = (ov > val) || (ov == val && oi < idx);
      val = take ? ov : val;
      idx = take ? oi : idx;
    }
    if (lane == 0) labels[(size_t)m * NTOK + base + v] = idx;
    else if (lane == 16) labels[(size_t)m * NTOK + base + v + 8] = idx;
  }
}

// ---------------------------------------------------------------------------
// Weighted member counts per centroid
// ---------------------------------------------------------------------------
__global__ __launch_bounds__(256) void k_count(const float* __restrict__ Wt,
                                               const int* __restrict__ labels,
                                               float* __restrict__ counts) {
  __shared__ float s[KCENT];
  int m = blockIdx.x;
  if (threadIdx.x < KCENT) s[threadIdx.x] = 0.0f;
  __syncthreads();
  for (int t = threadIdx.x; t < NTOK; t += 256) {
    float wv = Wt[(size_t)m * NTOK + t];
    if (wv > 0.0f) atomicAdd(&s[labels[(size_t)m * NTOK + t]], wv);
  }
  __syncthreads();
  if (threadIdx.x < KCENT) counts[m * KCENT + threadIdx.x] = s[threadIdx.x];
}

// ---------------------------------------------------------------------------
// Weighted sums: block = (mask, 128-channel slice, token quarter).
// Each thread exclusively owns one channel column -> plain LDS accumulate.
// ---------------------------------------------------------------------------
__global__ __launch_bounds__(128) void k_update(const float* __restrict__ X,
                                                const float* __restrict__ Wt,
                                                const int* __restrict__ labels,
                                                const int4* __restrict__ geo,
                                                float* __restrict__ sums) {
  __shared__ float s[KCENT * 128];
  int m = blockIdx.x >> 5;
  int r = blockIdx.x & 31;
  int cg = r >> 2, tgrp = r & 3;
  int c = cg * 128 + threadIdx.x;
  for (int j = threadIdx.x; j < KCENT * 128; j += 128) s[j] = 0.0f;
  __syncthreads();
  const float* xm = X + (size_t)m * FHH * FWW * CDIM;
  int t0 = tgrp * (NTOK / 4), t1 = t0 + (NTOK / 4);
  for (int t = t0; t < t1; t++) {
    float wv = Wt[(size_t)m * NTOK + t];
    if (wv == 0.0f) continue;  // uniform across block
    int l = labels[(size_t)m * NTOK + t];
    int y0, y1, x0, x1; float w00, w01, w10, w11;
    unpack_geo(geo[t], y0, y1, x0, x1, w00, w01, w10, w11);
    float v = w00 * xm[(size_t)(y0 * FWW + x0) * CDIM + c] +
              w01 * xm[(size_t)(y0 * FWW + x1) * CDIM + c] +
              w10 * xm[(size_t)(y1 * FWW + x0) * CDIM + c] +
              w11 * xm[(size_t)(y1 * FWW + x1) * CDIM + c];
    s[l * 128 + threadIdx.x] += wv * v;
  }
  __syncthreads();
#pragma unroll
  for (int j = 0; j < KCENT; j++)
    atomicAdd(&sums[((size_t)m * KCENT + j) * CDIM + c], s[j * 128 + threadIdx.x]);
}

// new = counts>0 ? sums / max(counts,1) : old
__global__ __launch_bounds__(256) void k_finalize(const float* __restrict__ sums,
                                                  const float* __restrict__ counts,
                                                  const float* __restrict__ centIn,
                                                  float* __restrict__ centOut) {
  int m = blockIdx.x / KCENT, j = blockIdx.x % KCENT;
  float cnt = counts[m * KCENT + j];
  float inv = 1.0f / fmaxf(cnt, 1.0f);
  size_t off = ((size_t)m * KCENT + j) * CDIM;
  for (int c = threadIdx.x; c < CDIM; c += 256)
    centOut[off + c] = (cnt > 0.0f) ? sums[off + c] * inv : centIn[off + c];
}

// ---------------------------------------------------------------------------
// GEMM: out(128 x 4096) = act(A(128 x K) @ B(4096 x K)^T + bias)
// One wave per 16x16 tile, WMMA f32 16x16x4 over K.
// ---------------------------------------------------------------------------
__global__ __launch_bounds__(256) void k_gemm(const float* __restrict__ A,
                                              const float* __restrict__ B,
                                              const float* __restrict__ bias,
                                              float* __restrict__ out, int K,
                                              int applyGelu) {
  int wave = threadIdx.x >> 5, lane = threadIdx.x & 31;
  int tileId = blockIdx.x * 8 + wave;   // 2048 tiles
  int nt = tileId & 255, mt = tileId >> 8;
  int rrow = lane & 15;
  const float* ap = A + (size_t)(mt * 16 + rrow) * K;
  const float* bp = B + (size_t)(nt * 16 + rrow) * K;
  int cOff = (lane < 16) ? 0 : 2;
  v8f acc = {0.f, 0.f, 0.f, 0.f, 0.f, 0.f, 0.f, 0.f};
  for (int k0 = 0; k0 < K; k0 += 4) {
    v2f a = *(const v2f*)(ap + k0 + cOff);
    v2f bv = *(const v2f*)(bp + k0 + cOff);
    acc = wmma_f32x4(a, bv, acc);
  }
  int col = nt * 16 + rrow;
  float bs = bias[col];
  int rbase = mt * 16 + ((lane < 16) ? 0 : 8);
#pragma unroll
  for (int v = 0; v < 8; v++) {
    float val = acc[v] + bs;
    if (applyGelu)
      val = 0.5f * val * (1.0f + erff(val * 0.70710678118654752f));
    out[(size_t)(rbase + v) * HIDD + col] = val;
  }
}

// ---------------------------------------------------------------------------
extern "C" void kernel_launch(void* const* d_in, const int* in_sizes, int n_in,
                              void* d_out, int out_size, void* d_ws,
                              size_t ws_size, hipStream_t stream) {
  const float* feats = (const float*)d_in[0];
  const float* masks = (const float*)d_in[1];
  const float* box_xy = (const float*)d_in[2];
  const float* raw_hw = (const float*)d_in[3];
  const float* Wp = (const float*)d_in[4];
  const float* bp = (const float*)d_in[5];
  const float* W1 = (const float*)d_in[6];
  const float* b1 = (const float*)d_in[7];
  const float* W2 = (const float*)d_in[8];
  const float* b2 = (const float*)d_in[9];

  char* ws = (char*)d_ws;
  float* X = (float*)(ws + 0);                      // 18,874,368 B
  int4* GEO = (int4*)(ws + 18874368);               //    200,704 B
  float* WT = (float*)(ws + 19075072);              //    401,408 B
  int* LAB = (int*)(ws + 19476480);                 //    401,408 B
  float* CA = (float*)(ws + 19877888);              //    524,288 B
  float* CB = (float*)(ws + 20402176);              //    524,288 B
  float* SUMS = (float*)(ws + 20926464);            //    524,288 B
  float* CNT = (float*)(ws + 21450752);             //        512 B
  int* IDX = (int*)(ws + 21451264);                 //        512 B
  float* H = (float*)(ws + 21451776);               //  2,097,152 B

  k_compute_x<<<MNUM * FHH * FWW, 256, 0, stream>>>(feats, box_xy, raw_hw, Wp,
                                                    bp, X);
  k_geo_w<<<NTOK / 256, 256, 0, stream>>>(masks, GEO, WT);
  k_init_select<<<MNUM, 32, 0, stream>>>(WT, IDX);
  k_gather_cent<<<MNUM * KCENT, 256, 0, stream>>>(X, GEO, IDX, CA);

  float* cin = CA;
  float* cout = CB;
  for (int it = 0; it < KM_ITERS; it++) {
    hipMemsetAsync(SUMS, 0, (size_t)MNUM * KCENT * CDIM * sizeof(float), stream);
    k_assign<<<MNUM * 98, 256, 0, stream>>>(X, cin, GEO, LAB);
    k_count<<<MNUM, 256, 0, stream>>>(WT, LAB, CNT);
    k_update<<<MNUM * 32, 128, 0, stream>>>(X, WT, LAB, GEO, SUMS);
    k_finalize<<<MNUM * KCENT, 256, 0, stream>>>(SUMS, CNT, cin, cout);
    float* tmp = cin; cin = cout; cout = tmp;
  }
  // after swap, `cin` holds the final centroids = mf (128 x 1024)

  k_gemm<<<256, 256, 0, stream>>>(cin, W1, b1, H, CDIM, 1);
  k_gemm<<<256, 256, 0, stream>>>(H, W2, b2, (float*)d_out, HIDD, 0);
}